// GCN_43654047596702
// MI455X (gfx1250) — compile-verified
//
#include <hip/hip_runtime.h>

// ---------------- problem constants (from reference) ----------------
#define NNODES 50000
#define NEDGES 1600000
#define F_IN   512
#define NHID   128
#define NCLS   64

typedef __attribute__((ext_vector_type(2))) float v2f;
typedef __attribute__((ext_vector_type(8))) float v8f;
typedef __attribute__((ext_vector_type(4))) int   v4i;

#define AS1 __attribute__((address_space(1)))
#define AS3 __attribute__((address_space(3)))

#if defined(__has_builtin)
#  if __has_builtin(__builtin_amdgcn_global_load_async_to_lds_b128) && \
      __has_builtin(__builtin_amdgcn_s_wait_asynccnt)
#    define HAVE_ASYNC_LDS 1
#  endif
#endif
#ifndef HAVE_ASYNC_LDS
#  define HAVE_ASYNC_LDS 0
#endif

// 16B global -> LDS tile copy: async DMA (ASYNCcnt) when available, sync fallback.
__device__ __forceinline__ void tile_ld16(const float* g, float* l) {
#if HAVE_ASYNC_LDS
    __builtin_amdgcn_global_load_async_to_lds_b128((AS1 v4i*)g, (AS3 v4i*)l, 0, 0);
#else
    *(float4*)l = *(const float4*)g;
#endif
}
__device__ __forceinline__ void wait_async0() {
#if HAVE_ASYNC_LDS
    __builtin_amdgcn_s_wait_asynccnt(0);
#endif
}

// ---------------- small utility kernels ----------------
__global__ __launch_bounds__(256) void zero_f32(float* __restrict__ p, size_t n) {
    size_t i = (size_t)blockIdx.x * 256 + threadIdx.x;
    if (i < n) p[i] = 0.0f;
}

__global__ __launch_bounds__(256) void degree_k(const int* __restrict__ dst,
                                                float* __restrict__ deg, int nE) {
    int e = blockIdx.x * 256 + threadIdx.x;
    if (e < nE) unsafeAtomicAdd(&deg[dst[e]], 1.0f);
}

__global__ __launch_bounds__(256) void dinv_k(const float* __restrict__ deg,
                                              float* __restrict__ dinv, int n) {
    int i = blockIdx.x * 256 + threadIdx.x;
    if (i < n) dinv[i] = rsqrtf(deg[i] + 1.0f);   // +1 for self loop
}

// W[K][N] -> Wt[N][K] so each lane's WMMA B fragment is one contiguous b64 load
__global__ __launch_bounds__(256) void transpose_w(const float* __restrict__ W,
                                                   float* __restrict__ Wt, int K, int N) {
    int i = blockIdx.x * 256 + threadIdx.x;
    if (i < K * N) {
        const int k = i / N, n = i % N;
        Wt[(size_t)n * K + k] = W[i];
    }
}

// ---------------- GEMM1: h1[N,128] = x[N,512] @ W1[512,128] (fp32 WMMA) ----------------
// 256 thr = 8 waves; block = 16-row strip x all 128 cols; wave w -> cols [16w,16w+16).
// K tiled by 64; x tile double-buffered in LDS via async global->LDS DMA.
__global__ __launch_bounds__(256) void gemm1_wmma(const float* __restrict__ x,
                                                  const float* __restrict__ Wt,   // [NHID][F_IN]
                                                  float* __restrict__ h) {
    __shared__ float tileA[2][16][68];            // pad 64->68: conflict-free, 16B aligned
    const int row0  = blockIdx.x * 16;
    const int tid   = threadIdx.x;
    const int wave  = tid >> 5;                   // 0..7
    const int lane  = tid & 31;
    const int col0  = wave * 16;
    const int mn    = lane & 15;                  // A row m == B/D col n
    const int khalf = (lane >> 4) << 1;           // K pair select per lane half

    const int r = tid >> 4;                       // loader: 0..15
    const int c = (tid & 15) << 2;                // 0..60 step 4
    const float* gA   = x  + (size_t)(row0 + r)  * F_IN + c;
    const float* Wrow = Wt + (size_t)(col0 + mn) * F_IN;

    tile_ld16(gA, &tileA[0][r][c]);               // prologue: tile 0

    v8f acc = {};
    constexpr int NT = F_IN / 64;                 // 8 K-tiles
    for (int t = 0; t < NT; ++t) {
        wait_async0();
        __syncthreads();                          // tile t complete & visible to all waves
        if (t + 1 < NT)
            tile_ld16(gA + (t + 1) * 64, &tileA[(t + 1) & 1][r][c]);
        const float (*tA)[68] = tileA[t & 1];
        const int kbase = t * 64;
        #pragma unroll
        for (int kk = 0; kk < 64; kk += 4) {
            v2f a, b;
            a.x = tA[mn][kk + khalf];             // A 16x4: VGPR0={K0|K2}, VGPR1={K1|K3}
            a.y = tA[mn][kk + khalf + 1];
            const float2 bv = *(const float2*)(Wrow + kbase + kk + khalf);
            b.x = bv.x; b.y = bv.y;               // B 4x16 mirrors A with N on lanes
            acc = __builtin_amdgcn_wmma_f32_16x16x4_f32(
                    false, a, false, b, (short)0, acc, false, false);
        }
    }
    #pragma unroll
    for (int rr = 0; rr < 8; ++rr) {              // D: VGPR rr -> row (rr | lanehalf*8)
        const int M = rr + ((lane >> 4) << 3);
        h[(size_t)(row0 + M) * NHID + col0 + mn] = acc[rr];
    }
}

// ---------------- GEMM2: h2[N,64] = h[N,128] @ W2[128,64] (fp32 WMMA) ----------------
// 128 thr = 4 waves (one per 16-col tile); K tiled by 32, double-buffered.
__global__ __launch_bounds__(128) void gemm2_wmma(const float* __restrict__ hIn,
                                                  const float* __restrict__ Wt,   // [NCLS][NHID]
                                                  float* __restrict__ o) {
    __shared__ float tileA[2][16][36];            // pad 32->36
    const int row0  = blockIdx.x * 16;
    const int tid   = threadIdx.x;
    const int wave  = tid >> 5;                   // 0..3
    const int lane  = tid & 31;
    const int col0  = wave * 16;
    const int mn    = lane & 15;
    const int khalf = (lane >> 4) << 1;

    const int r = tid >> 3;                       // loader: 0..15
    const int c = (tid & 7) << 2;                 // 0..28 step 4
    const float* gA   = hIn + (size_t)(row0 + r)  * NHID + c;
    const float* Wrow = Wt  + (size_t)(col0 + mn) * NHID;

    tile_ld16(gA, &tileA[0][r][c]);

    v8f acc = {};
    constexpr int NT = NHID / 32;                 // 4 K-tiles
    for (int t = 0; t < NT; ++t) {
        wait_async0();
        __syncthreads();
        if (t + 1 < NT)
            tile_ld16(gA + (t + 1) * 32, &tileA[(t + 1) & 1][r][c]);
        const float (*tA)[36] = tileA[t & 1];
        const int kbase = t * 32;
        #pragma unroll
        for (int kk = 0; kk < 32; kk += 4) {
            v2f a, b;
            a.x = tA[mn][kk + khalf];
            a.y = tA[mn][kk + khalf + 1];
            const float2 bv = *(const float2*)(Wrow + kbase + kk + khalf);
            b.x = bv.x; b.y = bv.y;
            acc = __builtin_amdgcn_wmma_f32_16x16x4_f32(
                    false, a, false, b, (short)0, acc, false, false);
        }
    }
    #pragma unroll
    for (int rr = 0; rr < 8; ++rr) {
        const int M = rr + ((lane >> 4) << 3);
        o[(size_t)(row0 + M) * NCLS + col0 + mn] = acc[rr];
    }
}

// ---------------- edge aggregation: agg[dst] += h[src] * dinv[src]*dinv[dst] ----------------
__global__ __launch_bounds__(256) void edge_agg128(const int* __restrict__ src,
                                                   const int* __restrict__ dst,
                                                   const float* __restrict__ dinv,
                                                   const float* __restrict__ h,
                                                   float* __restrict__ agg, int nE) {
    size_t g = (size_t)blockIdx.x * 256 + threadIdx.x;
    if (g >= (size_t)nE * 32) return;
    const int e = (int)(g >> 5);
    const int q = (int)(g & 31);
    const int s = src[e], d = dst[e];
    const float nrm = dinv[s] * dinv[d];
    const float4 v = ((const float4*)(h + (size_t)s * NHID))[q];
    float* out = agg + (size_t)d * NHID + (q << 2);
    unsafeAtomicAdd(out + 0, v.x * nrm);
    unsafeAtomicAdd(out + 1, v.y * nrm);
    unsafeAtomicAdd(out + 2, v.z * nrm);
    unsafeAtomicAdd(out + 3, v.w * nrm);
}

__global__ __launch_bounds__(256) void edge_agg64(const int* __restrict__ src,
                                                  const int* __restrict__ dst,
                                                  const float* __restrict__ dinv,
                                                  const float* __restrict__ h,
                                                  float* __restrict__ agg, int nE) {
    size_t g = (size_t)blockIdx.x * 256 + threadIdx.x;
    if (g >= (size_t)nE * 16) return;
    const int e = (int)(g >> 4);
    const int q = (int)(g & 15);
    const int s = src[e], d = dst[e];
    const float nrm = dinv[s] * dinv[d];
    const float4 v = ((const float4*)(h + (size_t)s * NCLS))[q];
    float* out = agg + (size_t)d * NCLS + (q << 2);
    unsafeAtomicAdd(out + 0, v.x * nrm);
    unsafeAtomicAdd(out + 1, v.y * nrm);
    unsafeAtomicAdd(out + 2, v.z * nrm);
    unsafeAtomicAdd(out + 3, v.w * nrm);
}

// ---------------- layer-1 epilogue: h = relu(agg + h1*dinv^2 + b1), in-place ----------------
__global__ __launch_bounds__(256) void post1_k(float* __restrict__ hb,
                                               const float* __restrict__ h1,
                                               const float* __restrict__ dinv,
                                               const float* __restrict__ b1) {
    size_t i = (size_t)blockIdx.x * 256 + threadIdx.x;
    if (i >= (size_t)NNODES * NHID) return;
    const int n = (int)(i >> 7);
    const int f = (int)(i & (NHID - 1));
    const float di = dinv[n];
    const float v = hb[i] + h1[i] * di * di + b1[f];
    hb[i] = v > 0.0f ? v : 0.0f;
}

// ---------------- layer-2 epilogue + softmax (one wave32 per row, 2 classes/lane) ----------------
__global__ __launch_bounds__(256) void post2_softmax(const float* __restrict__ agg,
                                                     const float* __restrict__ h2,
                                                     const float* __restrict__ dinv,
                                                     const float* __restrict__ b2,
                                                     float* __restrict__ out) {
    const int row  = blockIdx.x * 8 + (threadIdx.x >> 5);
    const int lane = threadIdx.x & 31;
    if (row >= NNODES) return;
    const float di = dinv[row];
    const float dd = di * di;
    const size_t base = (size_t)row * NCLS;
    float v0 = agg[base + lane]      + h2[base + lane]      * dd + b2[lane];
    float v1 = agg[base + lane + 32] + h2[base + lane + 32] * dd + b2[lane + 32];
    float mx = fmaxf(v0, v1);
    #pragma unroll
    for (int o = 16; o > 0; o >>= 1) mx = fmaxf(mx, __shfl_xor(mx, o, 32));
    const float e0 = expf(v0 - mx);
    const float e1 = expf(v1 - mx);
    float s = e0 + e1;
    #pragma unroll
    for (int o = 16; o > 0; o >>= 1) s += __shfl_xor(s, o, 32);
    const float inv = 1.0f / s;
    out[base + lane]      = e0 * inv;
    out[base + lane + 32] = e1 * inv;
}

// ---------------- host-side orchestration ----------------
extern "C" void kernel_launch(void* const* d_in, const int* in_sizes, int n_in,
                              void* d_out, int out_size, void* d_ws, size_t ws_size,
                              hipStream_t stream) {
    const float* x  = (const float*)d_in[0];
    const int*   ei = (const int*)  d_in[1];
    const float* W1 = (const float*)d_in[2];
    const float* b1 = (const float*)d_in[3];
    const float* W2 = (const float*)d_in[4];
    const float* b2 = (const float*)d_in[5];
    const int nE = in_sizes[1] / 2;                 // 1,600,000
    const int nN = in_sizes[0] / F_IN;              // 50,000
    const int* src = ei;
    const int* dst = ei + nE;

    const size_t n1 = (size_t)nN * NHID;
    const size_t n2 = (size_t)nN * NCLS;

    // workspace layout (floats)
    float* ws   = (float*)d_ws;
    float* deg  = ws;                               // nN
    float* dinv = deg  + 50176;                     // nN
    float* h1   = dinv + 50176;                     // nN*NHID
    float* hb   = h1   + n1;                        // nN*NHID (agg1 -> relu h, in place)
    float* h2   = hb   + n1;                        // nN*NCLS
    float* ag2  = h2   + n2;                        // nN*NCLS
    float* w1t  = ag2  + n2;                        // F_IN*NHID transposed W1
    float* w2t  = w1t  + (size_t)F_IN * NHID;       // NHID*NCLS transposed W2

    // re-zero accumulators every call (graph replay does not restore ws)
    zero_f32<<<(nN + 255) / 256, 256, 0, stream>>>(deg, (size_t)nN);
    zero_f32<<<(unsigned)((n1 + 255) / 256), 256, 0, stream>>>(hb, n1);
    zero_f32<<<(unsigned)((n2 + 255) / 256), 256, 0, stream>>>(ag2, n2);

    transpose_w<<<(F_IN * NHID + 255) / 256, 256, 0, stream>>>(W1, w1t, F_IN, NHID);
    transpose_w<<<(NHID * NCLS + 255) / 256, 256, 0, stream>>>(W2, w2t, NHID, NCLS);

    degree_k<<<(nE + 255) / 256, 256, 0, stream>>>(dst, deg, nE);
    dinv_k  <<<(nN + 255) / 256, 256, 0, stream>>>(deg, dinv, nN);

    // layer 1
    gemm1_wmma<<<nN / 16, 256, 0, stream>>>(x, w1t, h1);
    {
        size_t t = (size_t)nE * 32;
        edge_agg128<<<(unsigned)((t + 255) / 256), 256, 0, stream>>>(src, dst, dinv, h1, hb, nE);
    }
    post1_k<<<(unsigned)((n1 + 255) / 256), 256, 0, stream>>>(hb, h1, dinv, b1);

    // layer 2
    gemm2_wmma<<<nN / 16, 128, 0, stream>>>(hb, w2t, h2);
    {
        size_t t = (size_t)nE * 16;
        edge_agg64<<<(unsigned)((t + 255) / 256), 256, 0, stream>>>(src, dst, dinv, h2, ag2, nE);
    }
    post2_softmax<<<(nN + 7) / 8, 256, 0, stream>>>(ag2, h2, dinv, b2, (float*)d_out);
}